// GroupDROLoss_71717363908861
// MI455X (gfx1250) — compile-verified
//
#include <hip/hip_runtime.h>
#include <hip/hip_bf16.h>

// ---- CDNA5 vector types -----------------------------------------------------
typedef __attribute__((ext_vector_type(2))) float v2f;
typedef __attribute__((ext_vector_type(4))) float v4f;
typedef __attribute__((ext_vector_type(8))) float v8f;
typedef __attribute__((ext_vector_type(4))) int   v4i;

#define NUM_BINS 6
#define BLOCK 256
#define WAVES_PER_BLOCK (BLOCK / 32)
#define GRID_BLOCKS 4096

// Per-sample: CE = logsumexp(x,y,z) - logit[label]; bin = group*3 + label,
// group = (mean(logits) < 0.4) ? 0 : 1. Branchless 6-way accumulate.
__device__ __forceinline__ void accum_sample(float x, float y, float z, int lbl,
                                             float acc[NUM_BINS]) {
  float m = fmaxf(x, fmaxf(y, z));
  float s = __expf(x - m) + __expf(y - m) + __expf(z - m);
  float lse = m + __logf(s);
  float ly = (lbl == 0) ? x : ((lbl == 1) ? y : z);
  float ce = lse - ly;
  float br = (x + y + z) * (1.0f / 3.0f);
  int sg = ((br < 0.4f) ? 0 : 3) + lbl;
#pragma unroll
  for (int j = 0; j < NUM_BINS; ++j) acc[j] += (sg == j) ? ce : 0.0f;
}

__global__ void __launch_bounds__(BLOCK)
groupdro_partial_kernel(const float* __restrict__ logits,
                        const int* __restrict__ labels,
                        float* __restrict__ bins, int n) {
  float acc[NUM_BINS];
#pragma unroll
  for (int j = 0; j < NUM_BINS; ++j) acc[j] = 0.0f;

  const int gtid = blockIdx.x * BLOCK + threadIdx.x;
  const int stride = gridDim.x * BLOCK;
  const int chunks = n >> 2;  // 4 samples per chunk = 3 x b128 + 1 x b128

  const v4f* __restrict__ lp = reinterpret_cast<const v4f*>(logits);
  const v4i* __restrict__ lb = reinterpret_cast<const v4i*>(labels);

  for (int c = gtid; c < chunks; c += stride) {
    // streaming, single-use data: non-temporal loads keep L2 (192MB) clean
    v4f p0 = __builtin_nontemporal_load(&lp[3 * c + 0]);
    v4f p1 = __builtin_nontemporal_load(&lp[3 * c + 1]);
    v4f p2 = __builtin_nontemporal_load(&lp[3 * c + 2]);
    v4i l  = __builtin_nontemporal_load(&lb[c]);
    accum_sample(p0.x, p0.y, p0.z, l.x, acc);
    accum_sample(p0.w, p1.x, p1.y, l.y, acc);
    accum_sample(p1.z, p1.w, p2.x, l.z, acc);
    accum_sample(p2.y, p2.z, p2.w, l.w, acc);
  }
  // scalar tail (n % 4 leftovers)
  for (int i = (chunks << 2) + gtid; i < n; i += stride) {
    accum_sample(logits[3 * i], logits[3 * i + 1], logits[3 * i + 2],
                 labels[i], acc);
  }

  // ---- intra-wave butterfly reduction (wave32) ------------------------------
#pragma unroll
  for (int off = 16; off > 0; off >>= 1) {
#pragma unroll
    for (int j = 0; j < NUM_BINS; ++j)
      acc[j] += __shfl_xor(acc[j], off, 32);
  }

  __shared__ float wbins[WAVES_PER_BLOCK][NUM_BINS];
  const int lane = threadIdx.x & 31;
  const int wid = threadIdx.x >> 5;
  if (lane == 0) {
#pragma unroll
    for (int j = 0; j < NUM_BINS; ++j) wbins[wid][j] = acc[j];
  }
  __syncthreads();

  // ---- cross-wave reduction via V_WMMA_F32_16X16X4_F32 ----------------------
  // D = ones(16x4) x B(4x16) + C : every column of D becomes the sum of the 4
  // B values in that column (K permutation irrelevant since A is all ones).
  // Column n (< 6) carries bin n; the 4 K slots carry 4 waves' partials.
  // Two chained WMMAs cover all 8 waves. Wave-uniform branch => EXEC all-ones.
  if (wid == 0) {
    const int n16 = lane & 15;
    const int hi = lane >> 4;  // 0: lanes 0-15, 1: lanes 16-31
    float b0v = 0.0f, b1v = 0.0f, b2v = 0.0f, b3v = 0.0f;
    if (n16 < NUM_BINS) {
      const int w = hi * 2;           // waves {0,1} / {2,3}
      b0v = wbins[w][n16];
      b1v = wbins[w + 1][n16];
      b2v = wbins[w + 4][n16];        // waves {4,5} / {6,7}
      b3v = wbins[w + 5][n16];
    }
    v2f a;  a.x = 1.0f; a.y = 1.0f;   // all-ones A matrix
    v2f b0; b0.x = b0v; b0.y = b1v;
    v2f b1; b1.x = b2v; b1.y = b3v;
    v8f d = {};
    d = __builtin_amdgcn_wmma_f32_16x16x4_f32(false, a, false, b0,
                                              (short)0, d, false, false);
    d = __builtin_amdgcn_wmma_f32_16x16x4_f32(false, a, false, b1,
                                              (short)0, d, false, false);
    // D VGPR0, lanes 0-15 = row M=0, N=lane -> lane n holds bin-n block total
    if (hi == 0 && n16 < NUM_BINS) {
      atomicAdd(&bins[n16], d[0]);
    }
  }
}

__global__ void groupdro_zero_kernel(float* __restrict__ bins) {
  if (threadIdx.x < NUM_BINS) bins[threadIdx.x] = 0.0f;
}

__global__ void groupdro_finalize_kernel(const float* __restrict__ bins,
                                         const float* __restrict__ gw,
                                         float* __restrict__ out, float invN) {
  if (threadIdx.x == 0) {
    float s0 = bins[0], s1 = bins[1], s2 = bins[2];
    float s3 = bins[3], s4 = bins[4], s5 = bins[5];
    float g0 = s0 + s1 + s2;
    float g1 = s3 + s4 + s5;
    float total_loss = g0 * gw[0] + g1 * gw[1];
    float std_loss = (g0 + g1) * invN;
    out[0] = 0.7f * std_loss + 0.3f * total_loss;  // combined
    out[1] = g0; out[2] = g1;                      // group_losses
    out[3] = s0; out[4] = s1; out[5] = s2;         // subgroup_losses
    out[6] = s3; out[7] = s4; out[8] = s5;
  }
}

extern "C" void kernel_launch(void* const* d_in, const int* in_sizes, int n_in,
                              void* d_out, int out_size, void* d_ws, size_t ws_size,
                              hipStream_t stream) {
  const float* logits = (const float*)d_in[0];   // [N,3] f32
  const int* labels = (const int*)d_in[1];       // [N]   i32
  const float* gw = (const float*)d_in[2];       // [2]   f32
  float* out = (float*)d_out;                    // 9 floats
  float* bins = (float*)d_ws;                    // 6-float scratch
  const int n = in_sizes[1];

  groupdro_zero_kernel<<<1, 32, 0, stream>>>(bins);
  groupdro_partial_kernel<<<GRID_BLOCKS, BLOCK, 0, stream>>>(logits, labels,
                                                             bins, n);
  groupdro_finalize_kernel<<<1, 32, 0, stream>>>(bins, gw, out,
                                                 1.0f / (float)n);
}